// DeformConvImg2Col_10127532884580
// MI455X (gfx1250) — compile-verified
//
#include <hip/hip_runtime.h>

// Deformable-conv im2col for gfx1250 (MI455X).
// N=8, C=64, H=W=128, K=3x3, stride=1, pad=1, dil=1, DG=1, Ho=Wo=128.
//
// Memory-bound gather kernel: 302 MB streamed output (non-temporal stores so
// it does not thrash L2), data_im (33.5 MB) stays L2-resident for the ~1.2 GB
// of bilinear gather traffic. CDNA5 paths: Tensor Data Mover (tensor_load_to_lds
// + s_wait_tensorcnt) stages the strided offset/mask tiles into LDS; weights
// are computed once per (tap,pixel) and reused across all 64 channels.
// WMMA does not apply: gather addresses vary per output pixel, so no shared
// B operand exists and there is no matmul to tensorize.

#define KH 3
#define KW 3
#define KK 9
#define NN 8
#define CC 64
#define HH 128
#define WW 128
#define HO 128
#define WO 128
#define HW (HH * WW)              // 16384 elements per channel plane

#define USE_TDM 1

#if USE_TDM
typedef unsigned int v4u_ __attribute__((ext_vector_type(4)));
typedef int          v8i_ __attribute__((ext_vector_type(8)));
typedef int          v4i_ __attribute__((ext_vector_type(4)));

// Issue one TDM 2D tile load: `rows` rows of 128 contiguous fp32, row stride
// Ho*Wo elements, into LDS at byte offset lds_byte_off. D# per CDNA5 ISA §8.
__device__ static inline void tdm_load_rows(const void* gbase,
                                            unsigned lds_byte_off,
                                            unsigned rows) {
    const unsigned long long ga = (unsigned long long)gbase;
    v4u_ g0;
    g0.x = 1u;                                         // count=1, normal mode
    g0.y = lds_byte_off;                               // lds_addr
    g0.z = (unsigned)(ga & 0xffffffffu);               // global_addr[31:0]
    g0.w = (unsigned)((ga >> 32) & 0x1ffffffu)         // global_addr[56:32]
         | (2u << 30);                                 // type=2 ("image")
    v8i_ g1;
    g1[0] = (int)(2u << 16);          // data_size=2 (4 bytes); mask/flags=0
    g1[1] = (int)(128u << 16);        // tensor_dim0[15:0]=128 in bits[31:16]
    g1[2] = (int)(rows << 16);        // tensor_dim0 hi=0; tensor_dim1 lo=rows
    g1[3] = (int)(128u << 16);        // tensor_dim1 hi=0; tile_dim0=128
    g1[4] = (int)rows;                // tile_dim1=rows; tile_dim2=0 (2D)
    g1[5] = (int)(HO * WO);           // tensor_dim0_stride (elements) = 16384
    g1[6] = 0;                        // stride hi=0; tensor_dim1_stride lo=0
    g1[7] = 0;                        // tensor_dim1_stride hi=0
    const v4i_ z4 = {0, 0, 0, 0};
#if __clang_major__ >= 23
    const v8i_ z8 = {0, 0, 0, 0, 0, 0, 0, 0};
    __builtin_amdgcn_tensor_load_to_lds(g0, g1, z4, z4, z8, 0);
#else
    __builtin_amdgcn_tensor_load_to_lds(g0, g1, z4, z4, 0);
#endif
}
#endif

__global__ __launch_bounds__(256) void deform_im2col_kernel(
    const float* __restrict__ im,    // [N,C,H,W]
    const float* __restrict__ off,   // [N,2K,Ho,Wo]
    const float* __restrict__ msk,   // [N,K,Ho,Wo]
    float* __restrict__ out)         // [C*K,N,Ho,Wo]
{
    // Staged offset (18 rows) + mask (9 rows), each row = 128 floats.
    __shared__ float s_stage[27 * WO];       // 13824 B
    // Per-(tap,pixel) precomputed params, SoA for conflict-free reads.
    __shared__ float s_w[4][KK * WO];        // 18432 B
    __shared__ int   s_o[4][KK * WO];        // 18432 B

    const int tid = threadIdx.x;
    const int blk = blockIdx.x;
    const int n   = blk >> 7;        // blk / 128
    const int ho  = blk & 127;

    // ---------------- Phase A: stage offset+mask rows into LDS -------------
#if USE_TDM
    // One wave issues two TDM tile loads; TDM ignores EXEC, so guard with a
    // scalar (SGPR) branch to avoid duplicate issue from the other 7 waves.
    const unsigned waveid = __builtin_amdgcn_readfirstlane((unsigned)tid >> 5);
    if (waveid == 0) {
        const float* goff = off + ((size_t)(n * 18) * HO + ho) * WO;
        const float* gmsk = msk + ((size_t)(n * 9) * HO + ho) * WO;
        const unsigned lds0 = (unsigned)(unsigned long long)(&s_stage[0]);
        tdm_load_rows(goff, lds0, 18u);                     // 18x128 offsets
        tdm_load_rows(gmsk, lds0 + 18u * WO * 4u, 9u);      // 9x128 mask
        __builtin_amdgcn_s_wait_tensorcnt(0);
    }
    __syncthreads();
#else
    // Fallback: 27 rows * 32 chunks of 16B = 864 b128 async-to-LDS transfers.
    for (int j = tid; j < 27 * 32; j += 256) {
        const int r = j >> 5;        // row 0..26
        const int q = j & 31;        // 16B chunk within row
        const float* g;
        if (r < 18) {
            g = off + ((size_t)(n * 18 + r) * HO + ho) * WO + q * 4;
        } else {
            g = msk + ((size_t)(n * 9 + (r - 18)) * HO + ho) * WO + q * 4;
        }
        const unsigned lds_addr =
            (unsigned)(unsigned long long)(&s_stage[r * WO + q * 4]);
        const unsigned long long gaddr = (unsigned long long)g;
        asm volatile("global_load_async_to_lds_b128 %0, %1, off"
                     :: "v"(lds_addr), "v"(gaddr)
                     : "memory");
    }
    asm volatile("s_wait_asynccnt 0" ::: "memory");
    __syncthreads();
#endif

    // ---------------- Phase B: per-(tap,pixel) weights & offsets ------------
    for (int t = tid; t < KK * WO; t += 256) {
        const int k  = t >> 7;       // tap 0..8
        const int wo = t & 127;
        const float dy = s_stage[(2 * k) * WO + wo];
        const float dx = s_stage[(2 * k + 1) * WO + wo];
        const float m  = s_stage[18 * WO + k * WO + wo];
        const int ky = k / KW;
        const int kx = k - ky * KW;
        const float py = (float)(ho - 1 + ky) + dy;   // SH=DH=1, PH=1
        const float px = (float)(wo - 1 + kx) + dx;   // SW=DW=1, PW=1
        const float y0f = floorf(py);
        const float x0f = floorf(px);
        const float ly = py - y0f;
        const float lx = px - x0f;
        const int y0 = (int)y0f;
        const int x0 = (int)x0f;
        const int y1 = y0 + 1;
        const int x1 = x0 + 1;
        const float vy0 = (y0 >= 0 && y0 < HH) ? 1.0f : 0.0f;
        const float vy1 = (y1 >= 0 && y1 < HH) ? 1.0f : 0.0f;
        const float vx0 = (x0 >= 0 && x0 < WW) ? 1.0f : 0.0f;
        const float vx1 = (x1 >= 0 && x1 < WW) ? 1.0f : 0.0f;
        const float omy = 1.0f - ly;
        const float omx = 1.0f - lx;
        s_w[0][t] = m * omy * omx * vy0 * vx0;
        s_w[1][t] = m * omy * lx  * vy0 * vx1;
        s_w[2][t] = m * ly  * omx * vy1 * vx0;
        s_w[3][t] = m * ly  * lx  * vy1 * vx1;
        const int y0c = min(max(y0, 0), HH - 1);
        const int y1c = min(max(y1, 0), HH - 1);
        const int x0c = min(max(x0, 0), WW - 1);
        const int x1c = min(max(x1, 0), WW - 1);
        s_o[0][t] = y0c * WW + x0c;
        s_o[1][t] = y0c * WW + x1c;
        s_o[2][t] = y1c * WW + x0c;
        s_o[3][t] = y1c * WW + x1c;
    }
    __syncthreads();

    // ---------------- Phase C: apply over channels, coalesced NT stores -----
    const int wo  = tid & 127;
    const int ch0 = (tid >> 7) * (CC / 2);         // 0 or 32
    const float* imn = im + (size_t)n * CC * HW;
    const size_t outpix = (size_t)ho * WO + wo;
    const size_t cstride = (size_t)KK * NN * HO * WO;   // out stride per channel

    for (int k = 0; k < KK; ++k) {
        const int t = (k << 7) + wo;
        const float w00 = s_w[0][t];
        const float w01 = s_w[1][t];
        const float w10 = s_w[2][t];
        const float w11 = s_w[3][t];
        const int o00 = s_o[0][t];
        const int o01 = s_o[1][t];
        const int o10 = s_o[2][t];
        const int o11 = s_o[3][t];
        const float* p = imn + (size_t)ch0 * HW;
        float* op = out + ((size_t)(ch0 * KK + k) * NN + n) * (HO * WO) + outpix;
        #pragma unroll 4
        for (int c = 0; c < CC / 2; ++c) {
            float v = w00 * p[o00];
            v = fmaf(w01, p[o01], v);
            v = fmaf(w10, p[o10], v);
            v = fmaf(w11, p[o11], v);
            __builtin_nontemporal_store(v, op);   // keep L2 for the gathers
            p  += HW;
            op += cstride;
        }
    }
}

extern "C" void kernel_launch(void* const* d_in, const int* in_sizes, int n_in,
                              void* d_out, int out_size, void* d_ws, size_t ws_size,
                              hipStream_t stream) {
    (void)in_sizes; (void)n_in; (void)out_size; (void)d_ws; (void)ws_size;
    const float* im  = (const float*)d_in[0];
    const float* off = (const float*)d_in[1];
    const float* msk = (const float*)d_in[2];
    float* out = (float*)d_out;
    dim3 grid(NN * HO);   // 1024 blocks: one per (n, ho)
    dim3 block(256);      // 8 wave32
    hipLaunchKernelGGL(deform_im2col_kernel, grid, block, 0, stream,
                       im, off, msk, out);
}